// Bilaplacian_58763742544581
// MI455X (gfx1250) — compile-verified
//
#include <hip/hip_runtime.h>
#include <hip/hip_bf16.h>
#include <math.h>

// D=64, H=256. Directions: 64 (4e_i) + 4032 (3e_i+e_j, i!=j) + 2016 (2e_i+2e_j, i<j) = 6112.
// 6112 / 16 dirs-per-block = 382 blocks.
#define NDIR   6112
#define NBLK   382

typedef float v2f_t __attribute__((ext_vector_type(2)));
typedef float v8f_t __attribute__((ext_vector_type(8)));

// ---------------------------------------------------------------------------
// Kernel 1: shared jets state + W2 swizzled into WMMA-B per-lane layout.
//  T1[n] = tanh(W1 x + b1)[n]
//  T2[n] = tanh(W2 tanh(...) + b2)[n]
//  W2S[((t*64+s)*32+L)*2 + r] = W2[(t*16 + (L&15))*256 + s*4 + 2*(L>>4) + r]
//    -> for N-tile t, K-step s, lane L holds B-operand VGPR0/1 directly.
// ---------------------------------------------------------------------------
__global__ __launch_bounds__(256) void bilap_setup(
    const float* __restrict__ x,  const float* __restrict__ W1,
    const float* __restrict__ b1, const float* __restrict__ W2,
    const float* __restrict__ b2,
    float* __restrict__ T1, float* __restrict__ T2, float* __restrict__ W2S) {
  __shared__ float sh[256];
  const int tid = threadIdx.x;

  float a0 = b1[tid];
  for (int d = 0; d < 64; ++d) a0 += W1[tid * 64 + d] * x[d];
  float t1 = tanhf(a0);
  T1[tid] = t1;
  sh[tid] = t1;
  __syncthreads();

  float g0 = b2[tid];
  for (int c = 0; c < 256; ++c) g0 += W2[tid * 256 + c] * sh[c];
  T2[tid] = tanhf(g0);

  for (int idx = tid; idx < 65536; idx += 256) {
    const int r    = idx & 1;
    const int L    = (idx >> 1) & 31;
    const int s    = (idx >> 6) & 63;
    const int t    = idx >> 12;
    const int half = L >> 4;
    const int n    = L & 15;
    W2S[idx] = W2[(t * 16 + n) * 256 + s * 4 + 2 * half + r];
  }
}

// ---------------------------------------------------------------------------
// Kernel 2: per-block = 16 directions.
//  Phase 1: degree-4 tanh jets through layer 1 -> h1..h4 in LDS, stored in
//           WMMA A-operand layout (M rows = directions, 4 M-tiles = jet order).
//  Phase 2: G = H * W2^T via v_wmma_f32_16x16x4_f32 (K-loop of 64).
//  Phase 3: layer-2 tanh jet epilogue on accumulators + weighted reduction.
// ---------------------------------------------------------------------------
__global__ __launch_bounds__(256) void bilap_main(
    const float* __restrict__ W1, const float* __restrict__ T1,
    const float* __restrict__ T2, const float* __restrict__ W3,
    const float* __restrict__ W2S, float* __restrict__ partials) {
  extern __shared__ float smem[];
  float* HS   = smem;          // 16384 floats: A-operand (4 jet tiles x K=256)
  float* shT1 = smem + 16384;  // 256
  float* shT2 = shT1 + 256;    // 256
  float* shW3 = shT2 + 256;    // 256
  float* wts  = shW3 + 256;    // 16  (per-direction weight * 24)
  float* red  = wts + 16;      // 256

  const int tid = threadIdx.x;
  const int blk = blockIdx.x;

  shT1[tid] = T1[tid];
  shT2[tid] = T2[tid];
  shW3[tid] = W3[tid];

  // ---- decode direction dd = tid & 15 (matches reference enumeration) ----
  const int dd = tid & 15;
  const int g  = blk * 16 + dd;
  int i, j;
  float c1, c2, wt;
  if (g < 64) {                       // 4*e_i, weight alpha
    i = g; j = g; c1 = 4.0f; c2 = 0.0f;
    wt = 24.0f * (1.0f + 13.0f * 63.0f / 9.0f) / 256.0f;   // 24*alpha = 8.625
  } else if (g < 4096) {              // 3*e_i + e_j (i != j), weight beta
    const int t = g - 64;
    i = t / 63;
    const int jj = t % 63;
    j = jj + (jj >= i ? 1 : 0);
    c1 = 3.0f; c2 = 1.0f;
    wt = -24.0f / 36.0f;                                    // 24*beta
  } else {                            // 2*e_i + 2*e_j (i < j), weight gamma
    int rem = g - 4096;
    i = 0;
    while (rem >= 63 - i) { rem -= 63 - i; ++i; }
    j = i + 1 + rem;
    c1 = 2.0f; c2 = 2.0f;
    wt = 24.0f * 5.0f / 96.0f;                              // 24*gamma = 1.25
  }
  if (tid < 16) wts[tid] = wt;
  __syncthreads();   // shT1/wts visible

  // ---- Phase 1: layer-1 tanh jets (input jet: a0 shared, a1 = p, rest 0) --
  // y1 = p(1-y0^2); 2y2 = -p*w1; 3y3 = -p*w2; 4y4 = -p*w3, w = y^2 products.
  const int cbase = tid >> 4;          // 16 channels per thread
  for (int c = cbase; c < 256; c += 16) {
    const float p  = c1 * W1[c * 64 + i] + c2 * W1[c * 64 + j];
    const float y0 = shT1[c];
    const float s0 = 1.0f - y0 * y0;
    const float h1 = p * s0;
    const float h2 = -p * y0 * h1;
    const float h3 = -p * (2.0f * y0 * h2 + h1 * h1) * (1.0f / 3.0f);
    const float h4 = -p * (2.0f * y0 * h3 + 2.0f * h1 * h2) * 0.25f;
    // A-operand layout: tile (mt,s): lane = (k>>1)*16 + dir, vgpr r = k&1
    const int s    = c >> 2;
    const int kk   = c & 3;
    const int lane = (kk >> 1) * 16 + dd;
    const int r    = kk & 1;
    const int idx0 = (s * 32 + lane) * 2 + r;
    HS[idx0]         = h1;   // mt = 0
    HS[idx0 + 4096]  = h2;   // mt = 1
    HS[idx0 + 8192]  = h3;   // mt = 2
    HS[idx0 + 12288] = h4;   // mt = 3
  }
  __syncthreads();

  // ---- Phase 2: WMMA GEMM. wave owns N-tiles {2w, 2w+1} x all 4 M-tiles ---
  const int wave = tid >> 5;
  const int lane = tid & 31;
  const int nt0  = wave * 2;
  const int nt1  = nt0 + 1;

  const v8f_t zero = {0.f, 0.f, 0.f, 0.f, 0.f, 0.f, 0.f, 0.f};
  v8f_t acc[4][2];
#pragma unroll
  for (int m = 0; m < 4; ++m) { acc[m][0] = zero; acc[m][1] = zero; }

  const v2f_t* Ap = (const v2f_t*)HS;
  const v2f_t* Bp = (const v2f_t*)W2S;

  for (int s = 0; s < 64; ++s) {
    const v2f_t a0 = Ap[(0 * 64 + s) * 32 + lane];
    const v2f_t a1 = Ap[(1 * 64 + s) * 32 + lane];
    const v2f_t a2 = Ap[(2 * 64 + s) * 32 + lane];
    const v2f_t a3 = Ap[(3 * 64 + s) * 32 + lane];
    const v2f_t b0 = Bp[(nt0 * 64 + s) * 32 + lane];
    const v2f_t b1 = Bp[(nt1 * 64 + s) * 32 + lane];
    acc[0][0] = __builtin_amdgcn_wmma_f32_16x16x4_f32(false, a0, false, b0, (short)0, acc[0][0], false, false);
    acc[1][0] = __builtin_amdgcn_wmma_f32_16x16x4_f32(false, a1, false, b0, (short)0, acc[1][0], false, false);
    acc[2][0] = __builtin_amdgcn_wmma_f32_16x16x4_f32(false, a2, false, b0, (short)0, acc[2][0], false, false);
    acc[3][0] = __builtin_amdgcn_wmma_f32_16x16x4_f32(false, a3, false, b0, (short)0, acc[3][0], false, false);
    acc[0][1] = __builtin_amdgcn_wmma_f32_16x16x4_f32(false, a0, false, b1, (short)0, acc[0][1], false, false);
    acc[1][1] = __builtin_amdgcn_wmma_f32_16x16x4_f32(false, a1, false, b1, (short)0, acc[1][1], false, false);
    acc[2][1] = __builtin_amdgcn_wmma_f32_16x16x4_f32(false, a2, false, b1, (short)0, acc[2][1], false, false);
    acc[3][1] = __builtin_amdgcn_wmma_f32_16x16x4_f32(false, a3, false, b1, (short)0, acc[3][1], false, false);
  }

  // ---- Phase 3: layer-2 tanh jet epilogue on accumulators ----------------
  // D layout: lane -> neuron n = lane&15, half = lane>>4; VGPR v -> dir v+8*half
  const int n    = lane & 15;
  const int half = lane >> 4;
  float part = 0.0f;
#pragma unroll
  for (int q = 0; q < 2; ++q) {
    const int nn   = (nt0 + q) * 16 + n;
    const float y0 = shT2[nn];
    const float w3v = shW3[nn];
    const float s0 = 1.0f - y0 * y0;
#pragma unroll
    for (int v = 0; v < 8; ++v) {
      const float g1 = acc[0][q][v];
      const float g2 = acc[1][q][v];
      const float g3 = acc[2][q][v];
      const float g4 = acc[3][q][v];
      const float y1  = g1 * s0;
      const float w1  = 2.0f * y0 * y1;
      const float y2  = g2 * s0 - 0.5f * g1 * w1;
      const float w2  = 2.0f * y0 * y2 + y1 * y1;
      const float y3  = g3 * s0 - (2.0f * g2 * w1 + g1 * w2) * (1.0f / 3.0f);
      const float w3t = 2.0f * y0 * y3 + 2.0f * y1 * y2;
      const float y4  = g4 * s0 - (3.0f * g3 * w1 + 2.0f * g2 * w2 + g1 * w3t) * 0.25f;
      part += wts[v + 8 * half] * w3v * y4;   // wts already include the 24 = 4! factor
    }
  }

  red[tid] = part;
  __syncthreads();
  for (int off = 128; off > 0; off >>= 1) {
    if (tid < off) red[tid] += red[tid + off];
    __syncthreads();
  }
  if (tid == 0) partials[blk] = red[0];
}

// ---------------------------------------------------------------------------
// Kernel 3: deterministic fixed-order final reduction (no float atomics).
// ---------------------------------------------------------------------------
__global__ __launch_bounds__(256) void bilap_reduce(
    const float* __restrict__ partials, float* __restrict__ out) {
  __shared__ float red[256];
  const int tid = threadIdx.x;
  float v = 0.0f;
  for (int idx = tid; idx < NBLK; idx += 256) v += partials[idx];
  red[tid] = v;
  __syncthreads();
  for (int off = 128; off > 0; off >>= 1) {
    if (tid < off) red[tid] += red[tid + off];
    __syncthreads();
  }
  if (tid == 0) out[0] = red[0];
}

extern "C" void kernel_launch(void* const* d_in, const int* in_sizes, int n_in,
                              void* d_out, int out_size, void* d_ws, size_t ws_size,
                              hipStream_t stream) {
  (void)in_sizes; (void)n_in; (void)out_size; (void)ws_size;
  const float* x  = (const float*)d_in[0];   // (64,)
  const float* W1 = (const float*)d_in[1];   // (256, 64)
  const float* b1 = (const float*)d_in[2];   // (256,)
  const float* W2 = (const float*)d_in[3];   // (256, 256)
  const float* b2 = (const float*)d_in[4];   // (256,)
  const float* W3 = (const float*)d_in[5];   // (1, 256)
  // d_in[6] = b3: 4th derivative of constant is 0 -> unused.
  float* out = (float*)d_out;

  // workspace layout (floats): T1[256] | T2[256] | W2S[65536] | partials[382]
  // total = 66430 floats = 265,720 bytes
  float* ws       = (float*)d_ws;
  float* T1       = ws;
  float* T2       = ws + 256;
  float* W2S      = ws + 512;
  float* partials = ws + 512 + 65536;

  bilap_setup<<<1, 256, 0, stream>>>(x, W1, b1, W2, b2, T1, T2, W2S);

  // dyn LDS: HS 16384 + T1 256 + T2 256 + W3 256 + wts 16 + red 256 floats
  const size_t smem_bytes = (size_t)(16384 + 256 + 256 + 256 + 16 + 256) * sizeof(float);
  bilap_main<<<NBLK, 256, smem_bytes, stream>>>(W1, T1, T2, W3, W2S, partials);

  bilap_reduce<<<1, 256, 0, stream>>>(partials, out);
}